// AttnRNNModel_29635274342884
// MI455X (gfx1250) — compile-verified
//
#include <hip/hip_runtime.h>
#include <hip/hip_bf16.h>

// ---------------------------------------------------------------------------
// CDNA5 (gfx1250) AttnRNN seq2seq forward pass.
// All dense math via V_WMMA_F32_16X16X4_F32 (wave32, precision-true f32 path).
// B fragments use a k-pair-packed layout so every fragment is one b64 load.
// ---------------------------------------------------------------------------

typedef __attribute__((ext_vector_type(2))) float v2f;
typedef __attribute__((ext_vector_type(8))) float v8f;

__device__ __forceinline__ v8f wmma_f32(v2f a, v2f b, v8f c) {
  // (neg_a, A, neg_b, B, c_mod, C, reuse_a, reuse_b)
  return __builtin_amdgcn_wmma_f32_16x16x4_f32(false, a, false, b, (short)0, c,
                                               false, false);
}

// Model constants
#define MB   32     // batch
#define MS   128    // seq len
#define MEH  256    // encoder hidden
#define MDH  512    // decoder hidden

// ---------------------------------------------------------------------------
// Row gather:  out[r,:] = table[idx[r],:]
// ---------------------------------------------------------------------------
__global__ void gather_rows_kernel(const float* __restrict__ table,
                                   const int* __restrict__ idx,
                                   float* __restrict__ out, int width) {
  const int r = blockIdx.x;
  const float* src = table + (size_t)idx[r] * width;
  float* dst = out + (size_t)r * width;
  for (int c = threadIdx.x; c < width; c += blockDim.x) dst[c] = src[c];
}

// enc_hid[b,:] = Y[b, src_len[b]-1, :]
__global__ void gather_hid_kernel(const float* __restrict__ Y,
                                  const int* __restrict__ src_len,
                                  float* __restrict__ out) {
  const int b = blockIdx.x;
  const int t = src_len[b] - 1;
  for (int c = threadIdx.x; c < MDH; c += blockDim.x)
    out[(size_t)b * MDH + c] = Y[((size_t)b * MS + t) * MDH + c];
}

// ---------------------------------------------------------------------------
// Weight k-pair packing: P[(k>>1)*2N + 2n + (k&1)] = W[k*N + n]
// Makes every WMMA B-fragment ((k,n),(k+1,n)) a contiguous 8-byte load.
// ---------------------------------------------------------------------------
__global__ void pack_w_kernel(const float* __restrict__ W, float* __restrict__ P,
                              int K, int N) {
  size_t i = (size_t)blockIdx.x * blockDim.x + threadIdx.x;
  if (i >= (size_t)K * N) return;
  int k = (int)(i / N), n = (int)(i % N);
  P[(size_t)(k >> 1) * 2 * N + 2 * n + (k & 1)] = W[i];
}

// ---------------------------------------------------------------------------
// Generic LDS-staged WMMA GEMM:  C = act(A[MxK] @ B + bias)
//   transB==0: B row-major KxN.   transB==1: B row-major NxK (use B^T).
//   act: 0 = none, 1 = relu.
// Block = 256 threads (8 waves), tile 64x64, K-chunk 32.
// LDS A: pitch 34 (even -> aligned b64 frags, conflict-free strided reads).
// LDS B: k-pair packed, row stride 130.
// ---------------------------------------------------------------------------
#define BM  64
#define BN  64
#define BKC 32
#define APITCH 34
#define BPITCH (2 * BN + 2)   // 130

__global__ __launch_bounds__(256)
void gemm_wmma_kernel(const float* __restrict__ A, const float* __restrict__ Bm,
                      const float* __restrict__ bias, float* __restrict__ C,
                      int M, int N, int K, int transB, int act) {
  __shared__ float sA[BM * APITCH];
  __shared__ float sB[(BKC / 2) * BPITCH];
  const int tid  = threadIdx.x;
  const int lane = tid & 31, half = lane >> 4, l16 = lane & 15;
  const int wave = tid >> 5;
  const int row0 = blockIdx.y * BM;
  const int col0 = blockIdx.x * BN;
  const int tr  = (wave >> 1) * 16;
  const int tc0 = (wave & 1) * 32;
  const int tc1 = tc0 + 16;
  v8f acc0 = {}; v8f acc1 = {};

  for (int k0 = 0; k0 < K; k0 += BKC) {
#pragma unroll
    for (int i = 0; i < 8; ++i) {            // stage A tile (coalesced)
      int idx = tid + i * 256;
      int r = idx >> 5, c = idx & 31;
      int gr = row0 + r;
      sA[r * APITCH + c] = (gr < M) ? A[(size_t)gr * K + k0 + c] : 0.0f;
    }
    if (transB) {
#pragma unroll
      for (int i = 0; i < 8; ++i) {          // B^T rows, store k-pair packed
        int idx = tid + i * 256;
        int n = idx >> 5, kk = idx & 31;
        sB[(kk >> 1) * BPITCH + 2 * n + (kk & 1)] =
            Bm[(size_t)(col0 + n) * K + k0 + kk];
      }
    } else {
#pragma unroll
      for (int i = 0; i < 8; ++i) {
        int idx = tid + i * 256;
        int r = idx >> 6, c = idx & 63;
        sB[(r >> 1) * BPITCH + 2 * c + (r & 1)] =
            Bm[(size_t)(k0 + r) * N + col0 + c];
      }
    }
    __syncthreads();

    // prefetch next A K-chunk (gfx1250 global_prefetch)
    if (k0 + BKC < K) {
      int gr = row0 + wave * 8;
      if (gr < M) __builtin_prefetch(&A[(size_t)gr * K + k0 + BKC + lane], 0, 1);
    }

    // software pipeline: all fragment loads, then 16 back-to-back WMMAs
    v2f af[8], bf0[8], bf1[8];
#pragma unroll
    for (int u = 0; u < 8; ++u) {
      const int kc = 4 * u + 2 * half;       // even
      const int k2 = kc >> 1;
      af[u]  = *(const v2f*)&sA[(tr + l16) * APITCH + kc];
      bf0[u] = *(const v2f*)&sB[k2 * BPITCH + 2 * (tc0 + l16)];
      bf1[u] = *(const v2f*)&sB[k2 * BPITCH + 2 * (tc1 + l16)];
    }
#pragma unroll
    for (int u = 0; u < 8; ++u) {
      acc0 = wmma_f32(af[u], bf0[u], acc0);
      acc1 = wmma_f32(af[u], bf1[u], acc1);
    }
    __syncthreads();
  }

#pragma unroll
  for (int r = 0; r < 8; ++r) {
    int row = row0 + tr + r + 8 * half;
    if (row >= M) continue;
    int c0 = col0 + tc0 + l16;
    int c1 = col0 + tc1 + l16;
    float v0 = acc0[r], v1 = acc1[r];
    if (bias) { v0 += bias[c0]; v1 += bias[c1]; }
    if (act == 1) { v0 = fmaxf(v0, 0.0f); v1 = fmaxf(v1, 0.0f); }
    C[(size_t)row * N + c0] = v0;
    C[(size_t)row * N + c1] = v1;
  }
}

// ---------------------------------------------------------------------------
// Encoder RNN scan (grid.x = 2 directions, one WGP workgroup each).
//   h_t = tanh( G[dir][b,t,:] + h_{t-1} @ Whh[dir] + bhh[dir] )
// WhhP is k-pair packed -> B frags are single coalesced global b64 loads.
// h lives in LDS (pitch 258: even + conflict-free).
// ---------------------------------------------------------------------------
#define EHP (MEH + 2)   // 258

__global__ __launch_bounds__(256)
void enc_scan_kernel(const float* __restrict__ G,     // (2, B*S, 256)
                     const float* __restrict__ WhhP,  // (2, 128, 512) packed
                     const float* __restrict__ bhh,   // (2, 256)
                     float* __restrict__ Y) {         // (B*S, 512)
  const int H = MEH;
  __shared__ float hbuf[MB * EHP];
  const int dir = blockIdx.x;
  const float* Gd = G + (size_t)dir * MB * MS * H;
  const float* Wd = WhhP + (size_t)dir * H * H;
  const float* bd = bhh + (size_t)dir * H;
  const int tid  = threadIdx.x;
  const int lane = tid & 31, half = lane >> 4, l16 = lane & 15;
  const int wave = tid >> 5;
  const int c0 = wave * 32, c1 = c0 + 16;

  for (int i = tid; i < MB * EHP; i += 256) hbuf[i] = 0.0f;
  __syncthreads();

  for (int step = 0; step < MS; ++step) {
    const int t = dir ? (MS - 1 - step) : step;
    v8f acc00 = {}, acc01 = {}, acc10 = {}, acc11 = {};
    for (int k = 0; k < H; k += 4) {
      const int kc = k + 2 * half;           // even
      const int k2 = kc >> 1;
      v2f a0 = *(const v2f*)&hbuf[l16 * EHP + kc];
      v2f a1 = *(const v2f*)&hbuf[(16 + l16) * EHP + kc];
      v2f b0 = *(const v2f*)&Wd[(size_t)k2 * (2 * H) + 2 * (c0 + l16)];
      v2f b1 = *(const v2f*)&Wd[(size_t)k2 * (2 * H) + 2 * (c1 + l16)];
      acc00 = wmma_f32(a0, b0, acc00);
      acc01 = wmma_f32(a0, b1, acc01);
      acc10 = wmma_f32(a1, b0, acc10);
      acc11 = wmma_f32(a1, b1, acc11);
    }
    float v[4][8];
#pragma unroll
    for (int r = 0; r < 8; ++r) {
      const int b0r = r + 8 * half, b1r = 16 + r + 8 * half;
      const int cc0 = c0 + l16, cc1 = c1 + l16;
      v[0][r] = tanhf(acc00[r] + Gd[((size_t)b0r * MS + t) * H + cc0] + bd[cc0]);
      v[1][r] = tanhf(acc01[r] + Gd[((size_t)b0r * MS + t) * H + cc1] + bd[cc1]);
      v[2][r] = tanhf(acc10[r] + Gd[((size_t)b1r * MS + t) * H + cc0] + bd[cc0]);
      v[3][r] = tanhf(acc11[r] + Gd[((size_t)b1r * MS + t) * H + cc1] + bd[cc1]);
    }
    __syncthreads();                         // all h reads complete
#pragma unroll
    for (int r = 0; r < 8; ++r) {
      const int b0r = r + 8 * half, b1r = 16 + r + 8 * half;
      const int cc0 = c0 + l16, cc1 = c1 + l16;
      hbuf[b0r * EHP + cc0] = v[0][r];
      hbuf[b0r * EHP + cc1] = v[1][r];
      hbuf[b1r * EHP + cc0] = v[2][r];
      hbuf[b1r * EHP + cc1] = v[3][r];
      Y[((size_t)b0r * MS + t) * (2 * H) + dir * H + cc0] = v[0][r];
      Y[((size_t)b0r * MS + t) * (2 * H) + dir * H + cc1] = v[1][r];
      Y[((size_t)b1r * MS + t) * (2 * H) + dir * H + cc0] = v[2][r];
      Y[((size_t)b1r * MS + t) * (2 * H) + dir * H + cc1] = v[3][r];
    }
    __syncthreads();                         // h update visible
  }
}

// ---------------------------------------------------------------------------
// Decoder attention-RNN scan, one layer (1 block x 1024 threads = 32 waves).
//   score = softmax(h . enc_out);  attn = score @ enc_out
//   h = tanh((x_t + attn) @ Wih + bi + h @ Whh + bh)
// WihP/WhhP k-pair packed. Dynamic LDS: h(32x514), ax(32x514), prob, rsum.
// ---------------------------------------------------------------------------
#define DHP (MDH + 2)   // 514

__global__ __launch_bounds__(1024)
void dec_scan_kernel(const float* __restrict__ h0,    // (32,512)
                     const float* __restrict__ X,     // (B*S,512)
                     const float* __restrict__ encO,  // (B*S,512)
                     const float* __restrict__ WihP,  // (256,1024) packed
                     const float* __restrict__ WhhP,  // (256,1024) packed
                     const float* __restrict__ bi,    // (512)
                     const float* __restrict__ bh,    // (512)
                     float* __restrict__ Yout) {      // (B*S,512)
  extern __shared__ float sm[];
  float* h   = sm;                       // 32*514
  float* ax  = sm + MB * DHP;            // 32*514
  float* prb = ax + MB * DHP;            // 32*128
  float* rs  = prb + MB * MS;            // 32
  const int tid  = threadIdx.x;
  const int lane = tid & 31, half = lane >> 4, l16 = lane & 15;
  const int wave = tid >> 5;

  for (int i = tid; i < MB * MDH; i += 1024) {
    int b = i >> 9, c = i & 511;
    h[b * DHP + c] = h0[i];
  }
  __syncthreads();

  for (int t = 0; t < MS; ++t) {
    // ---- P1: raw scores (float4-vectorized dot products) ----------------
    {
      const int b = wave, j0 = lane;
      for (int i = 0; i < 4; ++i) {
        const int j = j0 + i * 32;
        const float4* er4 = (const float4*)(encO + ((size_t)b * MS + j) * MDH);
        float s = 0.0f;
        for (int c4 = 0; c4 < MDH / 4; ++c4) {
          float4 e = er4[c4];
          const float* hr = &h[b * DHP + 4 * c4];
          s += hr[0] * e.x + hr[1] * e.y + hr[2] * e.z + hr[3] * e.w;
        }
        prb[b * MS + j] = s;
      }
    }
    __syncthreads();
    // ---- P2: softmax (wave w owns batch w) -------------------------------
    {
      const int b = wave;
      float x0 = prb[b * MS + lane],      x1 = prb[b * MS + lane + 32];
      float x2 = prb[b * MS + lane + 64], x3 = prb[b * MS + lane + 96];
      float m = fmaxf(fmaxf(x0, x1), fmaxf(x2, x3));
      for (int o = 16; o > 0; o >>= 1) m = fmaxf(m, __shfl_xor(m, o, 32));
      float e0 = __expf(x0 - m), e1 = __expf(x1 - m);
      float e2 = __expf(x2 - m), e3 = __expf(x3 - m);
      float s = e0 + e1 + e2 + e3;
      for (int o = 16; o > 0; o >>= 1) s += __shfl_xor(s, o, 32);
      prb[b * MS + lane]      = e0; prb[b * MS + lane + 32] = e1;
      prb[b * MS + lane + 64] = e2; prb[b * MS + lane + 96] = e3;
      if (lane == 0) rs[b] = s;
    }
    __syncthreads();
    // ---- P3: ax = x_t + attn (lanes cover consecutive c: coalesced) -----
    {
      const int b = wave, cb = lane;
      const float inv = 1.0f / rs[b];
      const float* eb = encO + (size_t)b * MS * MDH;
      const float* xr = X + ((size_t)b * MS + t) * MDH;
      for (int i = 0; i < 16; ++i) {
        const int c = cb + i * 32;
        float a = 0.0f;
        for (int j = 0; j < MS; ++j) a += prb[b * MS + j] * eb[(size_t)j * MDH + c];
        ax[b * DHP + c] = xr[c] + a * inv;
      }
    }
    __syncthreads();
    // ---- P4: gate = ax@Wih + h@Whh (WMMA) --------------------------------
    v8f acc0 = {}, acc1 = {};
    {
      const int c0 = wave * 16;
      const int nn = 2 * (c0 + l16);
      for (int k = 0; k < MDH; k += 4) {
        const int kc = k + 2 * half;         // even
        const size_t k2 = (size_t)(kc >> 1);
        v2f aA0 = *(const v2f*)&ax[l16 * DHP + kc];
        v2f aA1 = *(const v2f*)&ax[(16 + l16) * DHP + kc];
        v2f aH0 = *(const v2f*)&h[l16 * DHP + kc];
        v2f aH1 = *(const v2f*)&h[(16 + l16) * DHP + kc];
        v2f bW = *(const v2f*)&WihP[k2 * (2 * MDH) + nn];
        v2f bV = *(const v2f*)&WhhP[k2 * (2 * MDH) + nn];
        acc0 = wmma_f32(aA0, bW, acc0);
        acc1 = wmma_f32(aA1, bW, acc1);
        acc0 = wmma_f32(aH0, bV, acc0);
        acc1 = wmma_f32(aH1, bV, acc1);
      }
    }
    __syncthreads();                       // all reads of h done
    // ---- P5: h update + output ------------------------------------------
    {
      const int cc = wave * 16 + l16;
      const float bsum = bi[cc] + bh[cc];
#pragma unroll
      for (int r = 0; r < 8; ++r) {
        const int b0r = r + 8 * half, b1r = 16 + r + 8 * half;
        const float v0 = tanhf(acc0[r] + bsum);
        const float v1 = tanhf(acc1[r] + bsum);
        h[b0r * DHP + cc] = v0;
        h[b1r * DHP + cc] = v1;
        Yout[((size_t)b0r * MS + t) * MDH + cc] = v0;
        Yout[((size_t)b1r * MS + t) * MDH + cc] = v1;
      }
    }
    __syncthreads();
  }
}

// ---------------------------------------------------------------------------
// Host-side orchestration
// ---------------------------------------------------------------------------
extern "C" void kernel_launch(void* const* d_in, const int* in_sizes, int n_in,
                              void* d_out, int out_size, void* d_ws, size_t ws_size,
                              hipStream_t stream) {
  const float* emb_enc    = (const float*)d_in[0];
  const float* enc_mlp_W1 = (const float*)d_in[1];
  const float* enc_mlp_b1 = (const float*)d_in[2];
  const float* enc_mlp_W2 = (const float*)d_in[3];
  const float* enc_mlp_b2 = (const float*)d_in[4];
  const float* enc_Wih0   = (const float*)d_in[5];
  const float* enc_Whh0   = (const float*)d_in[6];
  const float* enc_bih0   = (const float*)d_in[7];
  const float* enc_bhh0   = (const float*)d_in[8];
  const float* enc_Wih1   = (const float*)d_in[9];
  const float* enc_Whh1   = (const float*)d_in[10];
  const float* enc_bih1   = (const float*)d_in[11];
  const float* enc_bhh1   = (const float*)d_in[12];
  const float* emb_dec    = (const float*)d_in[13];
  const float* dmlp_W1    = (const float*)d_in[14];
  const float* dmlp_b1    = (const float*)d_in[15];
  const float* dmlp_W2    = (const float*)d_in[16];
  const float* dmlp_b2    = (const float*)d_in[17];
  const float* e2h_W1     = (const float*)d_in[18];
  const float* e2h_b1     = (const float*)d_in[19];
  const float* e2h_W2     = (const float*)d_in[20];
  const float* e2h_b2     = (const float*)d_in[21];
  const float* dec_Wih    = (const float*)d_in[22];
  const float* dec_Whh    = (const float*)d_in[23];
  const float* dec_bih    = (const float*)d_in[24];
  const float* dec_bhh    = (const float*)d_in[25];
  const float* h2e_W1     = (const float*)d_in[26];
  const float* h2e_b1     = (const float*)d_in[27];
  const float* h2e_W2     = (const float*)d_in[28];
  const float* h2e_b2     = (const float*)d_in[29];
  const int*   src        = (const int*)d_in[30];
  const int*   src_len    = (const int*)d_in[31];
  const int*   tgt        = (const int*)d_in[32];
  (void)in_sizes; (void)n_in; (void)out_size; (void)ws_size;

  float* ws = (float*)d_ws;
  size_t off = 0;
  auto wsalloc = [&](size_t n) { float* p = ws + off; off += n; return p; };
  const size_t NROW = (size_t)MB * MS;                 // 4096
  float* EA   = wsalloc(NROW * MEH);                   // gather / mlp ping
  float* EB   = wsalloc(NROW * MEH);                   // mlp pong
  float* G0   = wsalloc(2 * NROW * MEH);               // layer0 gates (both dirs)
  float* Y1   = wsalloc(NROW * 2 * MEH);               // layer0 output (B,S,512)
  float* G1   = wsalloc(2 * NROW * MEH);               // layer1 gates
  float* Y2   = wsalloc(NROW * 2 * MEH);               // enc_out (B,S,512)
  float* EHid = wsalloc((size_t)MB * 2 * MEH);         // enc_hid (32,512)
  float* H1   = wsalloc((size_t)MB * 1024);            // e2h hidden
  float* LH   = wsalloc((size_t)MB * 1024);            // lh flat (2,32,512)
  float* DA   = wsalloc(NROW * MDH);                   // decoder ping
  float* DB   = wsalloc(NROW * MDH);                   // decoder pong
  float* PW0  = wsalloc((size_t)2 * MEH * MEH);        // packed enc_Whh0
  float* PW1  = wsalloc((size_t)2 * MEH * MEH);        // packed enc_Whh1
  float* PDI  = wsalloc((size_t)2 * MDH * MDH);        // packed dec_Wih
  float* PDH  = wsalloc((size_t)2 * MDH * MDH);        // packed dec_Whh

  auto gemm = [&](const float* A, const float* Bm, const float* bias, float* C,
                  int M, int N, int K, int transB, int act) {
    dim3 grid(N / BN, (M + BM - 1) / BM);
    gemm_wmma_kernel<<<grid, 256, 0, stream>>>(A, Bm, bias, C, M, N, K, transB, act);
  };
  auto pack = [&](const float* W, float* P, int K, int N) {
    int total = K * N;
    pack_w_kernel<<<(total + 255) / 256, 256, 0, stream>>>(W, P, K, N);
  };

  // ---- weight packing for the recurrent scans ----
  pack(enc_Whh0,             PW0,             MEH, MEH);
  pack(enc_Whh0 + MEH * MEH, PW0 + MEH * MEH, MEH, MEH);
  pack(enc_Whh1,             PW1,             MEH, MEH);
  pack(enc_Whh1 + MEH * MEH, PW1 + MEH * MEH, MEH, MEH);
  pack(dec_Wih,              PDI,             MDH, MDH);
  pack(dec_Wih + MDH * MDH,  PDI + MDH * MDH, MDH, MDH);
  pack(dec_Whh,              PDH,             MDH, MDH);
  pack(dec_Whh + MDH * MDH,  PDH + MDH * MDH, MDH, MDH);

  // ---- encoder ----
  gather_rows_kernel<<<(int)NROW, 256, 0, stream>>>(emb_enc, src, EA, MEH);
  gemm(EA, enc_mlp_W1, enc_mlp_b1, EB, (int)NROW, MEH, MEH, 0, 1);
  gemm(EB, enc_mlp_W2, enc_mlp_b2, EA, (int)NROW, MEH, MEH, 0, 1);     // X
  gemm(EA, enc_Wih0,             enc_bih0,       G0,              (int)NROW, MEH, MEH, 0, 0);
  gemm(EA, enc_Wih0 + MEH * MEH, enc_bih0 + MEH, G0 + NROW * MEH, (int)NROW, MEH, MEH, 0, 0);
  enc_scan_kernel<<<2, 256, 0, stream>>>(G0, PW0, enc_bhh0, Y1);
  gemm(Y1, enc_Wih1,                 enc_bih1,       G1,              (int)NROW, MEH, 2 * MEH, 0, 0);
  gemm(Y1, enc_Wih1 + 2 * MEH * MEH, enc_bih1 + MEH, G1 + NROW * MEH, (int)NROW, MEH, 2 * MEH, 0, 0);
  enc_scan_kernel<<<2, 256, 0, stream>>>(G1, PW1, enc_bhh1, Y2);
  gather_hid_kernel<<<MB, 256, 0, stream>>>(Y2, src_len, EHid);

  // ---- enc_hid -> initial decoder hiddens (lh reshape is flat) ----
  gemm(EHid, e2h_W1, e2h_b1, H1, MB, 1024, 512, 0, 1);
  gemm(H1,   e2h_W2, e2h_b2, LH, MB, 1024, 1024, 0, 1);

  // ---- decoder input MLP ----
  gather_rows_kernel<<<(int)NROW, 256, 0, stream>>>(emb_dec, tgt, DA, MDH);
  gemm(DA, dmlp_W1, dmlp_b1, DB, (int)NROW, MDH, MDH, 0, 1);
  gemm(DB, dmlp_W2, dmlp_b2, DA, (int)NROW, MDH, MDH, 0, 1);           // ht

  // ---- decoder attention-RNN layers ----
  const size_t smbytes = (size_t)(2 * MB * DHP + MB * MS + MB) * sizeof(float);
  dec_scan_kernel<<<1, 1024, smbytes, stream>>>(
      LH, DA, Y2, PDI, PDH, dec_bih, dec_bhh, DB);
  dec_scan_kernel<<<1, 1024, smbytes, stream>>>(
      LH + MB * MDH, DB, Y2, PDI + MDH * MDH, PDH + MDH * MDH,
      dec_bih + MDH, dec_bhh + MDH, DA);

  // ---- h2e head ----
  gemm(DA, h2e_W1, h2e_b1, DB, (int)NROW, MDH, MDH, 0, 1);
  gemm(DB, h2e_W2, h2e_b2, DA, (int)NROW, MDH, MDH, 0, 0);             // hidden

  // ---- tied-embedding logits: hidden @ emb_dec^T  (4096 x 32000 x 512) ----
  gemm(DA, emb_dec, nullptr, (float*)d_out, (int)NROW, 32000, MDH, 1, 0);
}